// graphVectorKuramoto_69965017252366
// MI455X (gfx1250) — compile-verified
//
#include <hip/hip_runtime.h>
#include <math.h>

#define B_   8
#define H_   2048
#define KC_  1.0f
#define DT_  1.0f

typedef __attribute__((ext_vector_type(2))) float v2f;
typedef __attribute__((ext_vector_type(8))) float v8f;
typedef __attribute__((ext_vector_type(4))) int   v4i;

// ---- CDNA5 async global->LDS copy (ASYNCcnt) with portable fallback --------
#if __has_builtin(__builtin_amdgcn_global_load_async_to_lds_b128) && \
    __has_builtin(__builtin_amdgcn_s_wait_asynccnt)
#define HAS_ASYNC 1
#else
#define HAS_ASYNC 0
#endif

__device__ __forceinline__ void cp16(float* lds, const float* g) {
#if HAS_ASYNC
    // builtin signature (from hipcc diagnostic): param0 = v4i addrspace(1)*,
    // LDS side symmetric in addrspace(3); offset/cpol immediates.
    __builtin_amdgcn_global_load_async_to_lds_b128(
        (__attribute__((address_space(1))) v4i*)g,
        (__attribute__((address_space(3))) v4i*)lds, 0, 0);
#else
    *(float4*)lds = *(const float4*)g;
#endif
}
__device__ __forceinline__ void async_wait_all() {
#if HAS_ASYNC
    __builtin_amdgcn_s_wait_asynccnt(0);
#endif
}

// ---- gfx1250 hardware tanh if available ------------------------------------
__device__ __forceinline__ float fast_tanh(float x) {
#if __has_builtin(__builtin_amdgcn_tanhf)
    return __builtin_amdgcn_tanhf(x);
#else
    return tanhf(x);
#endif
}

// ---------------- pass 0: init per-batch min/max slots (uint-bit encoded) ----
__global__ void kur_init(unsigned* __restrict__ ws) {
    int t = threadIdx.x;
    if (t < 8)       ws[t] = 0x7F7FFFFFu;   // +FLT_MAX bits (min slot)
    else if (t < 16) ws[t] = 0u;            // 0.0f bits     (max slot)
}

// ---------------- pass 1: per-batch min/max of A_lat = relu(.5(A+A^T))+1e-6 --
__global__ __launch_bounds__(256) void kur_minmax(const float* __restrict__ A,
                                                  unsigned* __restrict__ ws) {
    const int b = blockIdx.x, i0 = blockIdx.y * 64, j0 = blockIdx.z * 64;
    const int t = threadIdx.x;
    __shared__ float sT[64][65];
    __shared__ float sMin[256], sMax[256];
    const float* Ab = A + (size_t)b * H_ * H_;
    #pragma unroll
    for (int it = 0; it < 16; ++it) {
        int idx = it * 256 + t;
        int jj = idx >> 6, ii = idx & 63;
        sT[jj][ii] = Ab[(size_t)(j0 + jj) * H_ + (i0 + ii)];
    }
    __syncthreads();
    float vmin = 3.4e38f, vmax = 0.f;
    #pragma unroll
    for (int it = 0; it < 16; ++it) {
        int idx = it * 256 + t;
        int r = idx >> 6, c = idx & 63;
        float v = fmaxf(0.5f * (Ab[(size_t)(i0 + r) * H_ + (j0 + c)] + sT[c][r]), 0.f) + 1e-6f;
        vmin = fminf(vmin, v);
        vmax = fmaxf(vmax, v);
    }
    sMin[t] = vmin; sMax[t] = vmax;
    __syncthreads();
    for (int s = 128; s > 0; s >>= 1) {
        if (t < s) { sMin[t] = fminf(sMin[t], sMin[t + s]); sMax[t] = fmaxf(sMax[t], sMax[t + s]); }
        __syncthreads();
    }
    if (t == 0) {
        atomicMin(&ws[b],     __float_as_uint(sMin[0]));
        atomicMax(&ws[8 + b], __float_as_uint(sMax[0]));
    }
}

// ---------------- pass 1.5: direction_mask = tanh(DL - DL^T), batch-invariant
__global__ __launch_bounds__(256) void kur_dmask(const float* __restrict__ DL,
                                                 float* __restrict__ dm) {
    int idx = blockIdx.x * 256 + threadIdx.x;
    int i = idx >> 11, j = idx & (H_ - 1);
    dm[idx] = fast_tanh(DL[(size_t)i * H_ + j] - DL[(size_t)j * H_ + i]);
}

// ---------------- pass 2: main Kuramoto step -------------------------------
// block = (b, 16-row i-tile), 256 threads = 8 waves.
// Double-buffered 64-wide j-tiles staged via GLOBAL_LOAD_ASYNC_TO_LDS_B128;
// V_WMMA_F32_16X16X4_F32 with B=ones performs the j-reduction on the matrix
// pipe while VALU/transcendental pipes compute rcp/tanh/sin.
// A-operand layout (ISA 7.12.2, 32-bit A 16x4): lane<16 -> M=lane, K=0(v0),1(v1);
// lane>=16 -> M=lane-16, K=2(v0),3(v1).
template <bool USE_DM>
__global__ __launch_bounds__(256) void kur_main(const float* __restrict__ theta,
                                                const float* __restrict__ gma,
                                                const float* __restrict__ A,
                                                const float* __restrict__ omega,
                                                const float* __restrict__ kappa,
                                                const float* __restrict__ DL,
                                                const float* __restrict__ DM,
                                                const unsigned* __restrict__ mm,
                                                float* __restrict__ out) {
    const int b    = blockIdx.x;
    const int i0   = blockIdx.y * 16;
    const int t    = threadIdx.x;
    const int lane = t & 31;
    const int wave = t >> 5;

    // strides 68/20 keep every 16B async store aligned AND the compute-phase
    // reads bank-conflict-free (68%64=4, 20 stride vs 16-lane runs).
    __shared__ __align__(16) float sArow[2][16][68];   // A rows
    __shared__ __align__(16) float sXrow[2][16][68];   // DM rows (USE_DM) or DL rows
    __shared__ __align__(16) float sAcolT[2][64][20];  // A transposed-source tile
    __shared__ __align__(16) float sDcolT[2][64][20];  // DL transposed tile (!USE_DM)
    __shared__ __align__(16) float sThJ[2][64][2];
    __shared__ float sThI[16][2];
    __shared__ float sPart[8][16][2];

    const float minA = __uint_as_float(mm[b]);
    const float maxA = __uint_as_float(mm[8 + b]);
    const float cmin = 1.f / maxA;
    const float cmax = 1.f / minA;
    const float invr = 1.f / (cmax - cmin + 1e-6f);

    const float* Ab = A + (size_t)b * H_ * H_;

    // per-thread copy mappings (one b128 per thread per tile array)
    const int rr = t >> 4, rc = (t & 15) << 2;   // 16 x 64 row tiles
    const int cj = t >> 2, ci = (t & 3) << 2;    // 64 x 16 col tiles

    auto issue_tile = [&](int bb, int jt) {
        cp16(&sArow[bb][rr][rc], &Ab[(size_t)(i0 + rr) * H_ + (jt + rc)]);
        if (USE_DM)
            cp16(&sXrow[bb][rr][rc], &DM[(size_t)(i0 + rr) * H_ + (jt + rc)]);
        else
            cp16(&sXrow[bb][rr][rc], &DL[(size_t)(i0 + rr) * H_ + (jt + rc)]);
        cp16(&sAcolT[bb][cj][ci], &Ab[(size_t)(jt + cj) * H_ + (i0 + ci)]);
        if (!USE_DM)
            cp16(&sDcolT[bb][cj][ci], &DL[(size_t)(jt + cj) * H_ + (i0 + ci)]);
        if (t < 128)
            sThJ[bb][t >> 1][t & 1] = theta[((size_t)b * H_ + jt + (t >> 1)) * 2 + (t & 1)];
    };

    if (t < 32) sThI[t >> 1][t & 1] = theta[((size_t)b * H_ + i0 + (t >> 1)) * 2 + (t & 1)];
    issue_tile(0, 0);
    async_wait_all();
    __syncthreads();

    const int ii = lane & 15;
    const int kb = (lane >> 4) << 1;          // K base: 0 or 2
    const float ti0 = sThI[ii][0];
    const float ti1 = sThI[ii][1];

    v8f acc0 = {};
    v8f acc1 = {};
    const v2f ones = {1.f, 1.f};

    int cur = 0;
    for (int jt = 0; jt < H_; jt += 64) {
        if (jt + 64 < H_) issue_tile(cur ^ 1, jt + 64);   // overlap with compute

        #pragma unroll
        for (int cc = 0; cc < 2; ++cc) {
            const int jc = 4 * (wave + 8 * cc);   // this wave's 4-wide j chunk
            const int ja = jc + kb;
            const int jb = ja + 1;

            float pa0, pa1, pb0, pb1;
            {
                float aij = sArow[cur][ii][ja], aji = sAcolT[cur][ja][ii];
                float al  = fmaxf(0.5f * (aij + aji), 0.f) + 1e-6f;
                float nc  = (__builtin_amdgcn_rcpf(al) - cmin) * invr;  // v_rcp_f32
                float dm  = USE_DM ? sXrow[cur][ii][ja]
                                   : fast_tanh(sXrow[cur][ii][ja] - sDcolT[cur][ja][ii]);
                float alpha = dm * nc;
                pa0 = al * __sinf(sThJ[cur][ja][0] - ti0 - alpha);
                pa1 = al * __sinf(sThJ[cur][ja][1] - ti1 - alpha);
            }
            {
                float aij = sArow[cur][ii][jb], aji = sAcolT[cur][jb][ii];
                float al  = fmaxf(0.5f * (aij + aji), 0.f) + 1e-6f;
                float nc  = (__builtin_amdgcn_rcpf(al) - cmin) * invr;
                float dm  = USE_DM ? sXrow[cur][ii][jb]
                                   : fast_tanh(sXrow[cur][ii][jb] - sDcolT[cur][jb][ii]);
                float alpha = dm * nc;
                pb0 = al * __sinf(sThJ[cur][jb][0] - ti0 - alpha);
                pb1 = al * __sinf(sThJ[cur][jb][1] - ti1 - alpha);
            }
            v2f a0 = {pa0, pb0};
            v2f a1 = {pa1, pb1};
            // D = A(16x4) * ones(4x16) + C : matrix pipe performs the j-reduction
            acc0 = __builtin_amdgcn_wmma_f32_16x16x4_f32(false, a0, false, ones,
                                                         (short)0, acc0, false, false);
            acc1 = __builtin_amdgcn_wmma_f32_16x16x4_f32(false, a1, false, ones,
                                                         (short)0, acc1, false, false);
        }

        async_wait_all();     // my async writes for the next buffer landed
        __syncthreads();      // all waves done reading 'cur' + next tile complete
        cur ^= 1;
    }

    // D layout: lane n<16 -> (M=v, N=n); lane n>=16 -> (M=v+8, N=n-16).
    // All columns equal (B=ones), so lanes 0 and 16 carry complete row sums.
    if (lane == 0 || lane == 16) {
        const int base = (lane >> 4) * 8;
        #pragma unroll
        for (int v = 0; v < 8; ++v) {
            sPart[wave][base + v][0] = acc0[v];
            sPart[wave][base + v][1] = acc1[v];
        }
    }
    __syncthreads();
    if (t < 32) {
        int i = t >> 1, d = t & 1;
        float s = 0.f;
        #pragma unroll
        for (int w = 0; w < 8; ++w) s += sPart[w][i][d];
        float th  = sThI[i][d];
        float gv  = gma[(size_t)b * H_ + i0 + i];
        float om  = omega[(i0 + i) * 2 + d];
        float kp  = kappa[(i0 + i) * 2 + d];
        float drv = kp * __sinf(gv - th);
        float dot = om + (KC_ / (float)H_) * s + drv;
        out[((size_t)b * H_ + i0 + i) * 2 + d] = th + DT_ * dot;
    }
}

extern "C" void kernel_launch(void* const* d_in, const int* in_sizes, int n_in,
                              void* d_out, int out_size, void* d_ws, size_t ws_size,
                              hipStream_t stream) {
    const float* theta = (const float*)d_in[0];
    const float* gma   = (const float*)d_in[1];
    const float* A     = (const float*)d_in[2];
    const float* omega = (const float*)d_in[3];
    const float* kappa = (const float*)d_in[4];
    const float* DL    = (const float*)d_in[5];
    float* out = (float*)d_out;

    unsigned* mm = (unsigned*)d_ws;
    float* dm = (float*)((char*)d_ws + 256);
    const size_t need = 256 + (size_t)H_ * H_ * sizeof(float);
    const bool useDM = (ws_size >= need);

    kur_init<<<dim3(1), dim3(32), 0, stream>>>(mm);
    kur_minmax<<<dim3(B_, H_ / 64, H_ / 64), dim3(256), 0, stream>>>(A, mm);
    if (useDM) {
        kur_dmask<<<dim3((H_ * H_) / 256), dim3(256), 0, stream>>>(DL, dm);
        kur_main<true><<<dim3(B_, H_ / 16), dim3(256), 0, stream>>>(
            theta, gma, A, omega, kappa, DL, dm, mm, out);
    } else {
        kur_main<false><<<dim3(B_, H_ / 16), dim3(256), 0, stream>>>(
            theta, gma, A, omega, kappa, DL, dm, mm, out);
    }
}